// SimpleGNN_43190191128704
// MI455X (gfx1250) — compile-verified
//
#include <hip/hip_runtime.h>

typedef __attribute__((ext_vector_type(2))) float v2f;
typedef __attribute__((ext_vector_type(8))) float v8f;

#define HIDF 64

// ---------------------------------------------------------------- utilities
__global__ void zero_f32(float* __restrict__ p, int n) {
    int i = blockIdx.x * blockDim.x + threadIdx.x;
    if (i < n) p[i] = 0.0f;
}

// deg[dst] += 1 per edge (deg buffer pre-zeroed)
__global__ void deg_count(const int* __restrict__ dst, float* __restrict__ deg, int nE) {
    int e = blockIdx.x * blockDim.x + threadIdx.x;
    if (e < nE) atomicAdd(&deg[dst[e]], 1.0f);
}

// in-place: dis = rsqrt(deg + 1)  (self-loop)
__global__ void deg_to_dis(float* __restrict__ deg, int n) {
    int i = blockIdx.x * blockDim.x + threadIdx.x;
    if (i < n) deg[i] = rsqrtf(deg[i] + 1.0f);
}

// ---------------------------------------------------------------- WMMA GEMM
// C[M x (NT*16)] = A[M x K] @ W[K x (NT*16)]  (+bias, +relu optional)
// One wave computes one 16-row tile across all NT column tiles.
// Uses V_WMMA_F32_16X16X4_F32: full-fp32 matrix core path.
template <int NT>
__launch_bounds__(128)
__global__ void gemm_wmma_f32(const float* __restrict__ A, const float* __restrict__ W,
                              const float* __restrict__ bias, float* __restrict__ C,
                              int M, int K, int doRelu) {
    const int lane = threadIdx.x & 31;
    const int wave = threadIdx.x >> 5;
    const int tile = blockIdx.x * (blockDim.x >> 5) + wave;   // wave-uniform
    if (tile * 16 >= M) return;                               // uniform exit: EXEC stays all-1s
    const int row0 = tile * 16;
    const int lid  = lane & 15;       // 0..15
    const int hi   = lane >> 4;       // lane group: 0 or 1
    const int N    = NT * 16;

    v8f acc[NT] = {};

    for (int k0 = 0; k0 < K; k0 += 4) {
        const int ka = k0 + hi * 2;
        // A fragment 16x4: lanes 0-15 rows, VGPR0/1 hold K = ka, ka+1
        v2f a;
        const float* ap = A + (size_t)(row0 + lid) * K + ka;
        a.x = ap[0];
        a.y = ap[1];
#pragma unroll
        for (int t = 0; t < NT; ++t) {
            // B fragment 4x16: rows ka (b.x) and ka+1 (b.y), N = lid within tile
            v2f b;
            const float* bp = W + (size_t)ka * N + t * 16 + lid;
            b.x = bp[0];
            b.y = bp[N];
            acc[t] = __builtin_amdgcn_wmma_f32_16x16x4_f32(
                false, a, false, b, (short)0, acc[t], false, false);
        }
    }

#pragma unroll
    for (int t = 0; t < NT; ++t) {
        const int col = t * 16 + lid;
        const float bv = bias ? bias[col] : 0.0f;
#pragma unroll
        for (int i = 0; i < 8; ++i) {
            const int r = row0 + hi * 8 + i;   // C/D layout: VGPR i -> M=i (lo) / M=8+i (hi)
            float v = acc[t][i] + bv;
            if (doRelu) v = fmaxf(v, 0.0f);
            if (r < M) C[(size_t)r * N + col] = v;
        }
    }
}

// ---------------------------------------------------------------- edge scatter
// agg[dst] += h[src] * dis[src]*dis[dst]; 16 threads/edge, float4 each
__global__ void edge_scatter(const float* __restrict__ h, const int* __restrict__ src,
                             const int* __restrict__ dst, const float* __restrict__ dis,
                             float* __restrict__ agg, int nEdges) {
    int t = blockIdx.x * blockDim.x + threadIdx.x;
    int e = t >> 4;
    if (e >= nEdges) return;
    int f = (t & 15) * 4;
    int s = src[e], d = dst[e];
    float norm = dis[s] * dis[d];
    float4 v = *(const float4*)(h + (size_t)s * HIDF + f);
    float* ap = agg + (size_t)d * HIDF + f;
    atomicAdd(ap + 0, v.x * norm);
    atomicAdd(ap + 1, v.y * norm);
    atomicAdd(ap + 2, v.z * norm);
    atomicAdd(ap + 3, v.w * norm);
}

// in-place: agg = relu(agg + h*dis^2 + bias)
__global__ void finalize_conv(float* __restrict__ agg, const float* __restrict__ h,
                              const float* __restrict__ dis, const float* __restrict__ bias,
                              int nNodes) {
    int t = blockIdx.x * blockDim.x + threadIdx.x;
    int n = t >> 4;
    if (n >= nNodes) return;
    int f = (t & 15) * 4;
    float dd = dis[n];
    float d2 = dd * dd;
    float4 a  = *(float4*)(agg + (size_t)n * HIDF + f);
    float4 hv = *(const float4*)(h + (size_t)n * HIDF + f);
    float4 bv = *(const float4*)(bias + f);
    a.x = fmaxf(a.x + hv.x * d2 + bv.x, 0.0f);
    a.y = fmaxf(a.y + hv.y * d2 + bv.y, 0.0f);
    a.z = fmaxf(a.z + hv.z * d2 + bv.z, 0.0f);
    a.w = fmaxf(a.w + hv.w * d2 + bv.w, 0.0f);
    *(float4*)(agg + (size_t)n * HIDF + f) = a;
}

// ---------------------------------------------------------------- mean pool
__global__ void pool_scatter(const float* __restrict__ h, const int* __restrict__ batch,
                             float* __restrict__ sums, float* __restrict__ cnt, int nNodes) {
    int t = blockIdx.x * blockDim.x + threadIdx.x;
    int n = t >> 4;
    if (n >= nNodes) return;
    int f = (t & 15) * 4;
    int g = batch[n];
    float4 v = *(const float4*)(h + (size_t)n * HIDF + f);
    float* sp = sums + (size_t)g * HIDF + f;
    atomicAdd(sp + 0, v.x);
    atomicAdd(sp + 1, v.y);
    atomicAdd(sp + 2, v.z);
    atomicAdd(sp + 3, v.w);
    if (f == 0) atomicAdd(&cnt[g], 1.0f);
}

__global__ void pool_div(float* __restrict__ sums, const float* __restrict__ cnt, int nGraphs) {
    int t = blockIdx.x * blockDim.x + threadIdx.x;
    if (t >= nGraphs * HIDF) return;
    sums[t] /= fmaxf(cnt[t >> 6], 1.0f);
}

// ---------------------------------------------------------------- final 32->2
__global__ void head2(const float* __restrict__ hid, const float* __restrict__ Wf2,
                      const float* __restrict__ bf2, float* __restrict__ out, int nGraphs) {
    int t = blockIdx.x * blockDim.x + threadIdx.x;
    if (t >= nGraphs * 2) return;
    int g = t >> 1, o = t & 1;
    float acc = bf2[o];
    const float* hp = hid + (size_t)g * 32;
#pragma unroll
    for (int k = 0; k < 32; ++k) acc += hp[k] * Wf2[k * 2 + o];
    out[t] = acc;
}

// ---------------------------------------------------------------- launch
static inline int cdiv(long long a, long long b) { return (int)((a + b - 1) / b); }

extern "C" void kernel_launch(void* const* d_in, const int* in_sizes, int n_in,
                              void* d_out, int out_size, void* d_ws, size_t ws_size,
                              hipStream_t stream) {
    const float* x     = (const float*)d_in[0];
    const int*   ei    = (const int*)d_in[1];
    const int*   batch = (const int*)d_in[2];
    const float* W1    = (const float*)d_in[3];
    const float* b1    = (const float*)d_in[4];
    const float* W2    = (const float*)d_in[5];
    const float* b2    = (const float*)d_in[6];
    const float* Wf1   = (const float*)d_in[7];
    const float* bf1   = (const float*)d_in[8];
    const float* Wf2   = (const float*)d_in[9];
    const float* bf2   = (const float*)d_in[10];
    float* out = (float*)d_out;

    const int nNodes  = in_sizes[0] / HIDF;
    const int nEdges  = in_sizes[1] / 2;
    const int nGraphs = out_size / 2;

    const int* src = ei;
    const int* dst = ei + nEdges;

    // workspace carve-up (floats)
    float* ws = (float*)d_ws;
    size_t off = 0;
    float* buf0 = ws + off; off += (size_t)nNodes * HIDF;
    float* buf1 = ws + off; off += (size_t)nNodes * HIDF;
    float* dis  = ws + off; off += ((size_t)nNodes + 63) & ~(size_t)63;
    float* sums = ws + off; off += (size_t)nGraphs * HIDF;
    float* cnt  = ws + off; off += (size_t)nGraphs;         // contiguous after sums
    float* hid  = ws + off; off += (size_t)nGraphs * 32;
    (void)ws_size; (void)n_in;

    const int ZB = 256;
    const long long nodeF  = (long long)nNodes * HIDF;
    const long long edgeT  = (long long)nEdges * 16;
    const long long nodeT  = (long long)nNodes * 16;

    // degree -> dis = rsqrt(deg+1)
    zero_f32<<<cdiv(nNodes, ZB), ZB, 0, stream>>>(dis, nNodes);
    deg_count<<<cdiv(nEdges, ZB), ZB, 0, stream>>>(dst, dis, nEdges);
    deg_to_dis<<<cdiv(nNodes, ZB), ZB, 0, stream>>>(dis, nNodes);

    const int gemmBlocks1 = cdiv(cdiv(nNodes, 16), 4);   // 4 waves per 128-thread block
    const int gemmBlocks2 = cdiv(cdiv(nGraphs, 16), 4);

    // ---- layer 1: h = x@W1 ; agg ; h1 = relu(agg + h*dis^2 + b1)
    gemm_wmma_f32<4><<<gemmBlocks1, 128, 0, stream>>>(x, W1, nullptr, buf0, nNodes, HIDF, 0);
    zero_f32<<<cdiv(nodeF, ZB), ZB, 0, stream>>>(buf1, (int)nodeF);
    edge_scatter<<<cdiv(edgeT, ZB), ZB, 0, stream>>>(buf0, src, dst, dis, buf1, nEdges);
    finalize_conv<<<cdiv(nodeT, ZB), ZB, 0, stream>>>(buf1, buf0, dis, b1, nNodes);

    // ---- layer 2: h = h1@W2 ; agg ; h2 = relu(agg + h*dis^2 + b2)
    gemm_wmma_f32<4><<<gemmBlocks1, 128, 0, stream>>>(buf1, W2, nullptr, buf0, nNodes, HIDF, 0);
    zero_f32<<<cdiv(nodeF, ZB), ZB, 0, stream>>>(buf1, (int)nodeF);
    edge_scatter<<<cdiv(edgeT, ZB), ZB, 0, stream>>>(buf0, src, dst, dis, buf1, nEdges);
    finalize_conv<<<cdiv(nodeT, ZB), ZB, 0, stream>>>(buf1, buf0, dis, b2, nNodes);

    // ---- mean pool over graphs (sums and cnt are contiguous: zero both at once)
    zero_f32<<<cdiv((long long)nGraphs * (HIDF + 1), ZB), ZB, 0, stream>>>(sums, nGraphs * (HIDF + 1));
    pool_scatter<<<cdiv(nodeT, ZB), ZB, 0, stream>>>(buf1, batch, sums, cnt, nNodes);
    pool_div<<<cdiv((long long)nGraphs * HIDF, ZB), ZB, 0, stream>>>(sums, cnt, nGraphs);

    // ---- head: hid = relu(g@Wf1 + bf1) [WMMA], out = hid@Wf2 + bf2
    gemm_wmma_f32<2><<<gemmBlocks2, 128, 0, stream>>>(sums, Wf1, bf1, hid, nGraphs, HIDF, 1);
    head2<<<cdiv((long long)nGraphs * 2, ZB), ZB, 0, stream>>>(hid, Wf2, bf2, out, nGraphs);
}